// AdditiveAttention_23390391894065
// MI455X (gfx1250) — compile-verified
//
#include <hip/hip_runtime.h>

typedef float v2f __attribute__((ext_vector_type(2)));
typedef float v8f __attribute__((ext_vector_type(8)));

constexpr int Bn = 8, Qn = 512, Kn = 512, Dn = 128, Hn = 128;
constexpr float NEG_INF = -1000000.0f;
constexpr float LOG2E   = 1.4426950408889634f;

// Native v_tanh_f32 on gfx1250 (TRANS op); fallback via v_exp_f32.
__device__ __forceinline__ float fast_tanh(float x) {
#if __has_builtin(__builtin_amdgcn_tanhf)
    return __builtin_amdgcn_tanhf(x);
#else
    // tanh(x) = 1 - 2/(2^(2x*log2e)+1); saturates correctly at +-1
    float e = __builtin_amdgcn_exp2f(x * (2.0f * LOG2E));
    return 1.0f - 2.0f / (e + 1.0f);
#endif
}

// ---------------------------------------------------------------------------
// Generic f32 GEMM via V_WMMA_F32_16X16X4_F32. One wave per 16x16 C tile.
// C[tr:tr+16, tc:tc+16] = sum_k A[tr:,k] * B[k,tc:]
// A layout (16x4, 2 VGPRs): lanes 0-15 hold K=0,1; lanes 16-31 hold K=2,3.
// B layout (4x16, 2 VGPRs): row-striped across lanes: v0 = rows {0,1}, v1 = rows {2,3}
//   split across lane halves.
// C/D layout (8 VGPRs): VGPR j -> row M=j (lanes 0-15) / M=j+8 (lanes 16-31).
// ---------------------------------------------------------------------------
__global__ void gemm_wmma_f32(const float* __restrict__ A, const float* __restrict__ B,
                              float* __restrict__ C, int inner, int lda, int ldb, int ldc,
                              long aBatchStride, long bBatchStride, long cBatchStride) {
    const float* Ab = A + (long)blockIdx.z * aBatchStride;
    const float* Bb = B + (long)blockIdx.z * bBatchStride;
    float*       Cb = C + (long)blockIdx.z * cBatchStride;

    const int tr   = blockIdx.y * 16;
    const int tc   = blockIdx.x * 16;
    const int lane = threadIdx.x & 31;
    const int half = lane >> 4;   // 0: lanes 0-15, 1: lanes 16-31
    const int lm   = lane & 15;

    v8f c = {};
    for (int kk = 0; kk < inner; kk += 4) {
        v2f a, b;
        const int ak = kk + half * 2;
        const float* ap = Ab + (long)(tr + lm) * lda + ak;
        a.x = ap[0];
        a.y = ap[1];
        const int bk = kk + half;
        b.x = Bb[(long)bk * ldb + tc + lm];
        b.y = Bb[(long)(bk + 2) * ldb + tc + lm];
        c = __builtin_amdgcn_wmma_f32_16x16x4_f32(false, a, false, b, (short)0, c,
                                                  false, false);
    }
#pragma unroll
    for (int j = 0; j < 8; ++j) {
        const int row = tr + j + half * 8;
        Cb[(long)row * ldc + tc + lm] = c[j];
    }
}

// ---------------------------------------------------------------------------
// scores[b,q,k] = sum_h w_v[h] * tanh(qp[b,q,h] + kp[b,k,h])
// One 256-thread block per 32x32 (q,k) tile per batch. qp/kp tiles staged in
// LDS (padded rows to dodge bank conflicts). Each thread owns 1 q and 4 k's.
// This is the dominant kernel: 268M tanh (TRANS) + 268M FMAs.
// ---------------------------------------------------------------------------
__global__ void scores_tanh_kernel(const float* __restrict__ qp, const float* __restrict__ kp,
                                   const float* __restrict__ wv, float* __restrict__ scores) {
    __shared__ __align__(16) float qs[32][Dn + 4];
    __shared__ __align__(16) float ks[32][Dn + 4];

    const int b   = blockIdx.z;
    const int q0  = blockIdx.y * 32;
    const int k0  = blockIdx.x * 32;
    const int tid = threadIdx.x;  // 256 threads

    const float* qpb = qp + ((long)b * Qn + q0) * Dn;
    const float* kpb = kp + ((long)b * Kn + k0) * Dn;

    // Stage 32x128 floats each (float4 loads): 1024 float4 per tile / 256 thr.
    for (int idx = tid; idx < 32 * (Dn / 4); idx += 256) {
        const int r  = idx >> 5;          // Dn/4 == 32 float4 per row
        const int c4 = (idx & 31) * 4;
        *(float4*)&qs[r][c4] = *(const float4*)&qpb[r * Dn + c4];
        *(float4*)&ks[r][c4] = *(const float4*)&kpb[r * Dn + c4];
    }
    __syncthreads();

    const int ql = tid >> 3;        // 0..31: local q row
    const int kb = (tid & 7) * 4;   // 0,4,...,28: first of 4 local k rows

    float acc0 = 0.f, acc1 = 0.f, acc2 = 0.f, acc3 = 0.f;
    for (int h = 0; h < Dn; h += 4) {
        const float4 qv  = *(const float4*)&qs[ql][h];
        const float4 w4  = *(const float4*)&wv[h];   // uniform -> scalar loads
        const float4 k0v = *(const float4*)&ks[kb + 0][h];
        const float4 k1v = *(const float4*)&ks[kb + 1][h];
        const float4 k2v = *(const float4*)&ks[kb + 2][h];
        const float4 k3v = *(const float4*)&ks[kb + 3][h];
        acc0 += w4.x * fast_tanh(qv.x + k0v.x) + w4.y * fast_tanh(qv.y + k0v.y)
              + w4.z * fast_tanh(qv.z + k0v.z) + w4.w * fast_tanh(qv.w + k0v.w);
        acc1 += w4.x * fast_tanh(qv.x + k1v.x) + w4.y * fast_tanh(qv.y + k1v.y)
              + w4.z * fast_tanh(qv.z + k1v.z) + w4.w * fast_tanh(qv.w + k1v.w);
        acc2 += w4.x * fast_tanh(qv.x + k2v.x) + w4.y * fast_tanh(qv.y + k2v.y)
              + w4.z * fast_tanh(qv.z + k2v.z) + w4.w * fast_tanh(qv.w + k2v.w);
        acc3 += w4.x * fast_tanh(qv.x + k3v.x) + w4.y * fast_tanh(qv.y + k3v.y)
              + w4.z * fast_tanh(qv.z + k3v.z) + w4.w * fast_tanh(qv.w + k3v.w);
    }

    float4 r4 = {acc0, acc1, acc2, acc3};
    *(float4*)(scores + ((long)b * Qn + q0 + ql) * Kn + k0 + kb) = r4;
}

// ---------------------------------------------------------------------------
// In-place masked softmax over K. One wave per row (wave32), 16 elems/lane,
// __shfl_xor reductions. Mask: k >= valid[b] -> NEG_INF before softmax
// (exp of ~-1e6 underflows to exactly 0, matching the reference).
// ---------------------------------------------------------------------------
__global__ void softmax_mask_kernel(float* __restrict__ attn, const int* __restrict__ valid) {
    const int row  = blockIdx.x * 8 + (threadIdx.x >> 5);
    const int lane = threadIdx.x & 31;
    const int b    = row / Qn;
    const int v    = valid[b];
    float* rp = attn + (long)row * Kn;

    float vals[16];
    float m = NEG_INF;
#pragma unroll
    for (int i = 0; i < 16; ++i) {
        const int k = i * 32 + lane;
        float s = rp[k];
        s = (k < v) ? s : NEG_INF;
        vals[i] = s;
        m = fmaxf(m, s);
    }
    for (int off = 16; off > 0; off >>= 1) m = fmaxf(m, __shfl_xor(m, off, 32));

    float sum = 0.f;
#pragma unroll
    for (int i = 0; i < 16; ++i) {
        const float e = __builtin_amdgcn_exp2f((vals[i] - m) * LOG2E);
        vals[i] = e;
        sum += e;
    }
    for (int off = 16; off > 0; off >>= 1) sum += __shfl_xor(sum, off, 32);

    const float inv = 1.0f / sum;
#pragma unroll
    for (int i = 0; i < 16; ++i) rp[i * 32 + lane] = vals[i] * inv;
}

extern "C" void kernel_launch(void* const* d_in, const int* in_sizes, int n_in,
                              void* d_out, int out_size, void* d_ws, size_t ws_size,
                              hipStream_t stream) {
    const float* queries = (const float*)d_in[0];  // (B,Q,D)
    const float* keys    = (const float*)d_in[1];  // (B,K,D)
    const float* values  = (const float*)d_in[2];  // (B,K,D)
    const int*   valid   = (const int*)d_in[3];    // (B,)
    const float* W_q     = (const float*)d_in[4];  // (D,H)
    const float* W_k     = (const float*)d_in[5];  // (D,H)
    const float* w_v     = (const float*)d_in[6];  // (H,)

    float* out  = (float*)d_out;                   // (B,Q,D)
    float* attn = out + (long)Bn * Qn * Dn;        // (B,Q,K)

    float* qp = (float*)d_ws;                      // (B*Q, H)  2 MB
    float* kp = qp + (long)Bn * Qn * Hn;           // (B*K, H)  2 MB

    // 1-2) q/k projections: flattened (B*Q) x D @ D x H, WMMA f32 16x16x4.
    dim3 gProj(Hn / 16, (Bn * Qn) / 16, 1);
    gemm_wmma_f32<<<gProj, 32, 0, stream>>>(queries, W_q, qp, Dn, Dn, Hn, Hn, 0, 0, 0);
    gemm_wmma_f32<<<gProj, 32, 0, stream>>>(keys,    W_k, kp, Dn, Dn, Hn, Hn, 0, 0, 0);

    // 3) additive-attention scores with tanh (the hot kernel), into attn buf.
    dim3 gS(Kn / 32, Qn / 32, Bn);
    scores_tanh_kernel<<<gS, 256, 0, stream>>>(qp, kp, w_v, attn);

    // 4) masked softmax over K, in place.
    softmax_mask_kernel<<<(Bn * Qn) / 8, 256, 0, stream>>>(attn, valid);

    // 5) output = attn @ values (batched), WMMA f32 16x16x4.
    dim3 gAV(Dn / 16, Qn / 16, Bn);
    gemm_wmma_f32<<<gAV, 32, 0, stream>>>(attn, values, out, Kn, Kn, Dn, Dn,
                                          (long)Qn * Kn, (long)Kn * Dn, (long)Qn * Dn);
}